// TFNConv_18631568130051
// MI455X (gfx1250) — compile-verified
//
#include <hip/hip_runtime.h>
#include <math.h>

typedef float v2f __attribute__((ext_vector_type(2)));
typedef float v8f __attribute__((ext_vector_type(8)));

#define MULT 32
#define NATTR 8
#define RB 16
#define FC 64
#define LOG2F_ 0.6931471805599453f

__device__ __forceinline__ float ssp(float x) {
    // numerically stable softplus(x) - log(2)
    return fmaxf(x, 0.0f) + log1pf(__expf(-fabsf(x))) - LOG2F_;
}

// ---------------------------------------------------------------- kernel 1
// linear_1: s = node_s @ W1s / sqrt(32);  v = einsum('nui,uw->nwi') / sqrt(32)
__global__ __launch_bounds__(256) void lin1_kernel(
    const float* __restrict__ node_s, const float* __restrict__ node_v,
    const float* __restrict__ W1s, const float* __restrict__ W1v,
    float* __restrict__ s_out, float* __restrict__ v_out, int N)
{
    __shared__ float Ls[MULT * MULT];
    __shared__ float Lv[MULT * MULT];
    for (int i = threadIdx.x; i < MULT * MULT; i += 256) { Ls[i] = W1s[i]; Lv[i] = W1v[i]; }
    __syncthreads();

    int gid = blockIdx.x * 256 + threadIdx.x;
    int n = gid >> 5, w = gid & 31;
    if (n >= N) return;
    const float inv_m = 0.17677669529663687f; // 1/sqrt(32)
    const float* srow = node_s + (size_t)n * MULT;
    const float* vrow = node_v + (size_t)n * MULT * 3;
    float as = 0.f, a0 = 0.f, a1 = 0.f, a2 = 0.f;
#pragma unroll 8
    for (int u = 0; u < MULT; ++u) {
        float ws_ = Ls[u * MULT + w];
        float wv_ = Lv[u * MULT + w];
        as += srow[u] * ws_;
        a0 += vrow[u * 3 + 0] * wv_;
        a1 += vrow[u * 3 + 1] * wv_;
        a2 += vrow[u * 3 + 2] * wv_;
    }
    s_out[(size_t)n * MULT + w] = as * inv_m;
    v_out[(size_t)n * MULT * 3 + w * 3 + 0] = a0 * inv_m;
    v_out[(size_t)n * MULT * 3 + w * 3 + 1] = a1 * inv_m;
    v_out[(size_t)n * MULT * 3 + w * 3 + 2] = a2 * inv_m;
}

// ---------------------------------------------------------------- kernel 2
// Radial MLP via V_WMMA_F32_16X16X4_F32 + CG tensor product + atomic scatter.
// 8 waves per block, 16 edges per wave -> 128 edges per block.
__global__ __launch_bounds__(256) void edge_kernel(
    const float* __restrict__ edge_emb, const float* __restrict__ edge_y1,
    const int* __restrict__ edge_index,
    const float* __restrict__ Wr1, const float* __restrict__ Wr2,
    const float* __restrict__ s_in, const float* __restrict__ v_in,
    float* __restrict__ agg_s, float* __restrict__ agg_v, int E)
{
    __shared__ float Wr1t[FC * RB];              // transposed [n][k]  64x16
    __shared__ float Wr2t[4 * MULT * FC];        // transposed [n][k] 128x64
    __shared__ float stage[8 * 16 * 4 * MULT];   // per-wave 16x128 staging

    const int tid = threadIdx.x;
    for (int i = tid; i < RB * FC; i += 256) {
        int k = i / FC, n = i % FC;
        Wr1t[n * RB + k] = Wr1[i];
    }
    for (int i = tid; i < FC * 4 * MULT; i += 256) {
        int k = i / (4 * MULT), n = i % (4 * MULT);
        Wr2t[n * FC + k] = Wr2[i];
    }
    __syncthreads();

    const int wave = tid >> 5, lane = tid & 31;
    const int m = lane & 15, kh = lane >> 4;     // tile row / K-half
    const long e0 = (long)blockIdx.x * 128 + wave * 16;
    float* const mystage = stage + wave * (16 * 4 * MULT);

    const v8f vzero = {0.f, 0.f, 0.f, 0.f, 0.f, 0.f, 0.f, 0.f};

    // ---- layer 1: h[16 x 64] = emb[16 x 16] @ Wr1, 4 N-tiles, K=16 in 4 steps
    v8f h[4];
#pragma unroll
    for (int t = 0; t < 4; ++t) h[t] = vzero;
#pragma unroll
    for (int k4 = 0; k4 < RB; k4 += 4) {
        long ee = e0 + m; if (ee >= E) ee = E - 1;
        v2f a = *(const v2f*)(edge_emb + ee * RB + k4 + 2 * kh);
#pragma unroll
        for (int t = 0; t < 4; ++t) {
            v2f b = *(const v2f*)(Wr1t + (t * 16 + m) * RB + k4 + 2 * kh);
            h[t] = __builtin_amdgcn_wmma_f32_16x16x4_f32(
                false, a, false, b, (short)0, h[t], false, false);
        }
    }
    // scale 1/sqrt(16), ssp, stash to LDS in row-major [edge_row][hid]
#pragma unroll
    for (int t = 0; t < 4; ++t)
#pragma unroll
        for (int r = 0; r < 8; ++r) {
            int row = r + 8 * kh, col = t * 16 + m;   // C/D layout
            mystage[row * FC + col] = ssp(h[t][r] * 0.25f);
        }
    __syncthreads();

    // ---- layer 2: w[16 x 128] = h[16 x 64] @ Wr2, 8 N-tiles, K=64 in 16 steps
    v8f wacc[8];
#pragma unroll
    for (int t = 0; t < 8; ++t) wacc[t] = vzero;
    for (int k4 = 0; k4 < FC; k4 += 4) {
        v2f a = *(const v2f*)(mystage + m * FC + k4 + 2 * kh);
#pragma unroll
        for (int t = 0; t < 8; ++t) {
            v2f b = *(const v2f*)(Wr2t + (t * 16 + m) * FC + k4 + 2 * kh);
            wacc[t] = __builtin_amdgcn_wmma_f32_16x16x4_f32(
                false, a, false, b, (short)0, wacc[t], false, false);
        }
    }
    __syncthreads();
#pragma unroll
    for (int t = 0; t < 8; ++t)
#pragma unroll
        for (int r = 0; r < 8; ++r) {
            int row = r + 8 * kh, col = t * 16 + m;
            mystage[row * 4 * MULT + col] = wacc[t][r] * 0.125f; // 1/sqrt(64)
        }
    __syncthreads();

    // ---- phase 2: CG 'uvu' tensor product + scatter (y0 == 1 constant)
    // thread t handles edge tid>>1, u-half (tid&1)*16
    long e = (long)blockIdx.x * 128 + (tid >> 1);
    if (e < E) {
        int wv = (tid >> 1) >> 4, row = (tid >> 1) & 15;
        const float* wrow = stage + wv * (16 * 4 * MULT) + row * 4 * MULT;
        int src = edge_index[e];
        int dst = edge_index[(long)E + e];
        float y0 = edge_y1[e * 3 + 0];
        float y1 = edge_y1[e * 3 + 1];
        float y2 = edge_y1[e * 3 + 2];
        const float inv_s3  = 0.5773502691896258f;  // 1/sqrt(3)
        const float inv_deg = 0.17677669529663687f; // 1/sqrt(32)
        const float* srow = s_in + (size_t)src * MULT;
        const float* vrow = v_in + (size_t)src * MULT * 3;
        float* as_ = agg_s + (size_t)dst * 2 * MULT;
        float* av_ = agg_v + (size_t)dst * 2 * MULT * 3;
        int u0 = (tid & 1) * 16;
#pragma unroll 4
        for (int u = u0; u < u0 + 16; ++u) {
            float w1 = wrow[u];
            float w2 = wrow[MULT + u];
            float w3 = wrow[2 * MULT + u];
            float w4 = wrow[3 * MULT + u];
            float se = srow[u];
            float v0 = vrow[u * 3 + 0], v1 = vrow[u * 3 + 1], v2 = vrow[u * 3 + 2];
            // 0e x 0e -> 0e
            unsafeAtomicAdd(as_ + u, w1 * se * inv_deg);
            // 1o x 1o -> 0e
            float dv = v0 * y0 + v1 * y1 + v2 * y2;
            unsafeAtomicAdd(as_ + MULT + u, w4 * dv * inv_s3 * inv_deg);
            // 0e x 1o -> 1o
            float t2 = w2 * se * inv_deg;
            unsafeAtomicAdd(av_ + u * 3 + 0, t2 * y0);
            unsafeAtomicAdd(av_ + u * 3 + 1, t2 * y1);
            unsafeAtomicAdd(av_ + u * 3 + 2, t2 * y2);
            // 1o x 0e -> 1o
            float t3 = w3 * inv_deg;
            unsafeAtomicAdd(av_ + 3 * MULT + u * 3 + 0, t3 * v0);
            unsafeAtomicAdd(av_ + 3 * MULT + u * 3 + 1, t3 * v1);
            unsafeAtomicAdd(av_ + 3 * MULT + u * 3 + 2, t3 * v2);
        }
    }
}

// ---------------------------------------------------------------- kernel 3
// linear_2 + self-connection, write final [N,128] output
__global__ __launch_bounds__(256) void out_kernel(
    const float* __restrict__ node_s, const float* __restrict__ node_v,
    const float* __restrict__ node_attrs,
    const float* __restrict__ W2s, const float* __restrict__ W2v,
    const float* __restrict__ Wscs, const float* __restrict__ Wscv,
    const float* __restrict__ agg_s, const float* __restrict__ agg_v,
    float* __restrict__ out, int N)
{
    __shared__ float L2s[2 * MULT * MULT];
    __shared__ float L2v[2 * MULT * MULT];
    __shared__ float Lss[MULT * NATTR * MULT];
    __shared__ float Lsv[MULT * NATTR * MULT];
    for (int i = threadIdx.x; i < 2 * MULT * MULT; i += 256) { L2s[i] = W2s[i]; L2v[i] = W2v[i]; }
    for (int i = threadIdx.x; i < MULT * NATTR * MULT; i += 256) { Lss[i] = Wscs[i]; Lsv[i] = Wscv[i]; }
    __syncthreads();

    int gid = blockIdx.x * 256 + threadIdx.x;
    int n = gid >> 5, w = gid & 31;
    if (n >= N) return;
    const float inv2m = 0.125f;   // 1/sqrt(64)
    const float invfan = 0.0625f; // 1/sqrt(32*8)

    const float* ags = agg_s + (size_t)n * 2 * MULT;
    const float* agv = agg_v + (size_t)n * 2 * MULT * 3;
    float ns = 0.f, nv0 = 0.f, nv1 = 0.f, nv2 = 0.f;
#pragma unroll 8
    for (int u = 0; u < 2 * MULT; ++u) {
        float w2s = L2s[u * MULT + w];
        float w2v = L2v[u * MULT + w];
        ns  += ags[u] * w2s;
        nv0 += agv[u * 3 + 0] * w2v;
        nv1 += agv[u * 3 + 1] * w2v;
        nv2 += agv[u * 3 + 2] * w2v;
    }
    ns *= inv2m; nv0 *= inv2m; nv1 *= inv2m; nv2 *= inv2m;

    float attr[NATTR];
#pragma unroll
    for (int a = 0; a < NATTR; ++a) attr[a] = node_attrs[(size_t)n * NATTR + a];

    const float* srow = node_s + (size_t)n * MULT;
    const float* vrow = node_v + (size_t)n * MULT * 3;
    float scs = 0.f, scv0 = 0.f, scv1 = 0.f, scv2 = 0.f;
#pragma unroll 4
    for (int u = 0; u < MULT; ++u) {
        float ts = 0.f, tv = 0.f;
#pragma unroll
        for (int a = 0; a < NATTR; ++a) {
            ts += attr[a] * Lss[(u * NATTR + a) * MULT + w];
            tv += attr[a] * Lsv[(u * NATTR + a) * MULT + w];
        }
        scs  += srow[u] * ts;
        scv0 += vrow[u * 3 + 0] * tv;
        scv1 += vrow[u * 3 + 1] * tv;
        scv2 += vrow[u * 3 + 2] * tv;
    }
    scs *= invfan; scv0 *= invfan; scv1 *= invfan; scv2 *= invfan;

    float* orow = out + (size_t)n * 4 * MULT;
    orow[w] = ns + scs;
    orow[MULT + w * 3 + 0] = nv0 + scv0;
    orow[MULT + w * 3 + 1] = nv1 + scv1;
    orow[MULT + w * 3 + 2] = nv2 + scv2;
}

// ---------------------------------------------------------------- launcher
extern "C" void kernel_launch(void* const* d_in, const int* in_sizes, int n_in,
                              void* d_out, int out_size, void* d_ws, size_t ws_size,
                              hipStream_t stream) {
    const float* node_s     = (const float*)d_in[0];
    const float* node_v     = (const float*)d_in[1];
    const float* node_attrs = (const float*)d_in[2];
    const float* edge_emb   = (const float*)d_in[3];
    // d_in[4] = edge_y0 (all-ones, constant) -- folded out analytically
    const float* edge_y1    = (const float*)d_in[5];
    const int*   edge_index = (const int*)d_in[6];
    const float* W1s  = (const float*)d_in[7];
    const float* W1v  = (const float*)d_in[8];
    const float* Wr1  = (const float*)d_in[9];
    const float* Wr2  = (const float*)d_in[10];
    const float* W2s  = (const float*)d_in[11];
    const float* W2v  = (const float*)d_in[12];
    const float* Wscs = (const float*)d_in[13];
    const float* Wscv = (const float*)d_in[14];

    const int N = in_sizes[0] / MULT;
    const int E = in_sizes[3] / RB;

    float* s_buf = (float*)d_ws;                       // N*32
    float* v_buf = s_buf + (size_t)N * MULT;           // N*96
    float* agg_s = v_buf + (size_t)N * MULT * 3;       // N*64
    float* agg_v = agg_s + (size_t)N * 2 * MULT;       // N*192

    hipMemsetAsync(agg_s, 0, (size_t)N * 8 * MULT * sizeof(float), stream);

    lin1_kernel<<<(N * MULT + 255) / 256, 256, 0, stream>>>(
        node_s, node_v, W1s, W1v, s_buf, v_buf, N);
    edge_kernel<<<(E + 127) / 128, 256, 0, stream>>>(
        edge_emb, edge_y1, edge_index, Wr1, Wr2, s_buf, v_buf, agg_s, agg_v, E);
    out_kernel<<<(N * MULT + 255) / 256, 256, 0, stream>>>(
        node_s, node_v, node_attrs, W2s, W2v, Wscs, Wscv, agg_s, agg_v,
        (float*)d_out, N);
}